// GMMSampler_33217277067380
// MI455X (gfx1250) — compile-verified
//
#include <hip/hip_runtime.h>
#include <hip/hip_bf16.h>

#define B_ 32
#define S_ 128
#define C_ 64
#define D_ 512
#define INV_TAU 1.0f

typedef __attribute__((ext_vector_type(2))) float v2f;
typedef __attribute__((ext_vector_type(4))) float f4;
typedef __attribute__((ext_vector_type(8))) float v8f;

// ---------------------------------------------------------------------------
// Kernel 1: w[b,s,:] = softmax((log(alpha[b,:]) + gumbel[b,s,:]) / tau)
// One wave32 per (b,s) row; lane holds c = lane and c = lane+32.
// ---------------------------------------------------------------------------
__global__ __launch_bounds__(256) void gmm_softmax_w(
    const float* __restrict__ alpha, const float* __restrict__ gumbel,
    float* __restrict__ w)
{
    const int lane = threadIdx.x & 31;
    const int row  = blockIdx.x * 8 + (threadIdx.x >> 5);   // b*S + s
    const int b    = row >> 7;                              // S = 128

    float x0 = (logf(alpha[b * C_ + lane])      + gumbel[(size_t)row * C_ + lane])      * INV_TAU;
    float x1 = (logf(alpha[b * C_ + lane + 32]) + gumbel[(size_t)row * C_ + lane + 32]) * INV_TAU;

    float m = fmaxf(x0, x1);
#pragma unroll
    for (int off = 16; off; off >>= 1) m = fmaxf(m, __shfl_xor(m, off, 32));

    float e0 = expf(x0 - m);
    float e1 = expf(x1 - m);
    float s  = e0 + e1;
#pragma unroll
    for (int off = 16; off; off >>= 1) s += __shfl_xor(s, off, 32);

    const float inv = 1.0f / s;
    w[(size_t)row * C_ + lane]      = e0 * inv;
    w[(size_t)row * C_ + lane + 32] = e1 * inv;
}

// ---------------------------------------------------------------------------
// Kernel 2: z[b,s,d] = sum_c w[b,s,c] * mu[b,c,d]   (exact fp32 WMMA GEMM)
// One wave per 16x16 (s,d) tile; K loop of 16 x v_wmma_f32_16x16x4_f32.
// A (16x4 f32): lane holds M = lane%16; VGPR0 K = (lane<16?0:2), VGPR1 K = +1.
// B (4x16 f32): lane holds N = lane%16; VGPR0 K = (lane<16?0:2), VGPR1 K = +1.
// C/D (16x16 f32): VGPR i -> M = i + (lane<16?0:8), N = lane%16.
// ---------------------------------------------------------------------------
__global__ __launch_bounds__(128) void gmm_wmu_wmma(
    const float* __restrict__ w, const float* __restrict__ mu,
    float* __restrict__ z)
{
    const int lane = threadIdx.x & 31;
    const int tile = blockIdx.x * 4 + (threadIdx.x >> 5);   // 0 .. 8191
    const int dt   = tile & 31;          // 32 d-tiles
    const int st   = (tile >> 5) & 7;    // 8 s-tiles
    const int b    = tile >> 8;          // 32 batches
    const int s0   = st * 16;
    const int d0   = dt * 16;
    const int half = lane >> 4;          // 0: lanes 0-15, 1: lanes 16-31
    const int l16  = lane & 15;

    const float* wrow = w  + ((size_t)b * S_ + s0 + l16) * C_;        // A row M=l16
    const float* mcol = mu + (size_t)b * C_ * D_ + d0 + l16;          // B col N=l16

    v8f acc = {};
#pragma unroll
    for (int k = 0; k < C_; k += 4) {
        const int kk = k + half * 2;
        v2f a, bb;
        a[0]  = wrow[kk];
        a[1]  = wrow[kk + 1];
        bb[0] = mcol[(size_t)kk * D_];
        bb[1] = mcol[(size_t)(kk + 1) * D_];
        acc = __builtin_amdgcn_wmma_f32_16x16x4_f32(
            /*neg_a=*/false, a, /*neg_b=*/false, bb,
            /*c_mod=*/(short)0, acc, /*reuse_a=*/false, /*reuse_b=*/false);
    }

    float* zp = z + ((size_t)b * S_ + s0 + half * 8) * D_ + d0 + l16;
#pragma unroll
    for (int i = 0; i < 8; ++i)
        zp[(size_t)i * D_] = acc[i];
}

// ---------------------------------------------------------------------------
// Kernel 3: z[b,s,d] += sum_c (w[b,s,c] * std[b,c,d]) * eps[b,s,c,d]
// One 128-thread block per (b,s); 4 contiguous d's per thread (b128 loads).
// eps is a 512 MiB one-shot stream -> non-temporal loads keep mu/std in L2.
// ---------------------------------------------------------------------------
__global__ __launch_bounds__(128) void gmm_eps_stream(
    const float* __restrict__ w, const float* __restrict__ stdv,
    const float* __restrict__ eps, float* __restrict__ z)
{
    const int row = blockIdx.x;          // b*S + s
    const int b   = row >> 7;

    __shared__ float ws[C_];
    if (threadIdx.x < C_) ws[threadIdx.x] = w[(size_t)row * C_ + threadIdx.x];
    __syncthreads();

    const int d = threadIdx.x * 4;
    const f4* sp = (const f4*)(stdv + (size_t)b   * C_ * D_ + d);  // stride D_/4 per c
    const f4* ep = (const f4*)(eps  + (size_t)row * C_ * D_ + d);

    f4 acc = {0.0f, 0.0f, 0.0f, 0.0f};
#pragma unroll 4
    for (int c = 0; c < C_; ++c) {
        const float wc = ws[c];
        f4 sv = sp[(size_t)c * (D_ / 4)];                          // L2-resident, reused over S
        f4 ev = __builtin_nontemporal_load(ep + (size_t)c * (D_ / 4)); // stream once
        acc += (wc * sv) * ev;
    }

    f4* zp = (f4*)(z + (size_t)row * D_ + d);
    f4 prev = *zp;       // written by gmm_wmu_wmma earlier on the same stream
    *zp = prev + acc;
}

// ---------------------------------------------------------------------------
extern "C" void kernel_launch(void* const* d_in, const int* in_sizes, int n_in,
                              void* d_out, int out_size, void* d_ws, size_t ws_size,
                              hipStream_t stream) {
    (void)in_sizes; (void)n_in; (void)out_size; (void)ws_size;
    const float* alpha  = (const float*)d_in[0];   // [B,C]
    const float* mu     = (const float*)d_in[1];   // [B,C,D]
    const float* stdv   = (const float*)d_in[2];   // [B,C,D]
    const float* gumbel = (const float*)d_in[3];   // [B,S,C]
    const float* eps    = (const float*)d_in[4];   // [B,S,C,D]
    float* z = (float*)d_out;                      // [B,S,D]
    float* w = (float*)d_ws;                       // [B,S,C] = 1 MiB scratch

    // 1) softmax weights: 4096 rows, 8 rows (waves) per 256-thread block
    gmm_softmax_w<<<(B_ * S_) / 8, 256, 0, stream>>>(alpha, gumbel, w);

    // 2) z = w @ mu via fp32 WMMA: 8192 tiles, 4 waves per 128-thread block
    gmm_wmu_wmma<<<(B_ * (S_ / 16) * (D_ / 16)) / 4, 128, 0, stream>>>(w, mu, z);

    // 3) z += (w*std)*eps streamed reduction: one block per (b,s)
    gmm_eps_stream<<<B_ * S_, 128, 0, stream>>>(w, stdv, eps, z);
}